// QuantumNATQuantum_65481071404259
// MI455X (gfx1250) — compile-verified
//
#include <hip/hip_runtime.h>

typedef __attribute__((ext_vector_type(2))) float v2f;
typedef __attribute__((ext_vector_type(4))) float v4f;
typedef __attribute__((ext_vector_type(8))) float v8f;

#define DIMQ   128
#define LDSROW 132          // pad 128->132 words: conflict-free b64 frag reads
#define WAVES  8
#define TPW    2            // 16-row tiles per wave
#define ROWS_PER_BLOCK (WAVES * TPW * 16)   // 256

// ---------------------------------------------------------------------------
// Kernel 1: build the 128x128 circuit operator U from theta.
// One block, 1024 threads: thread = (column c = tid&127, row-worker w = tid>>7).
// State lives in LDS (padded rows, conflict-free: lanes vary in c -> banks
// consecutive). Per RY gate the 64 butterfly pairs are split across the 8
// row-workers; barriers between gates. Rolled loops -> tiny static code,
// no register spills.
// ---------------------------------------------------------------------------
__global__ void __launch_bounds__(1024)
qnat_build_u(const float* __restrict__ theta, float* __restrict__ Uout) {
  extern __shared__ float Ub[];                 // 128 x LDSROW
  const int c = threadIdx.x & 127;
  const int w = threadIdx.x >> 7;               // 0..7

  // init U = I ; worker w owns rows [16w, 16w+16)
  for (int i = 16 * w; i < 16 * w + 16; ++i)
    Ub[i * LDSROW + c] = (i == c) ? 1.0f : 0.0f;
  __syncthreads();

  for (int g = 0; g < 14; ++g) {
    const int q = (g >= 7) ? (g - 7) : g;
    const float half = 0.5f * theta[g];
    const float cc = cosf(half);
    const float sn = sinf(half);
    const int lo = 1 << q;

    // worker w handles pairs p in [8w, 8w+8)
    for (int p = 8 * w; p < 8 * w + 8; ++p) {
      const int i0 = ((p >> q) << (q + 1)) | (p & (lo - 1));
      const int i1 = i0 + lo;
      const float a0 = Ub[i0 * LDSROW + c];
      const float a1 = Ub[i1 * LDSROW + c];
      Ub[i0 * LDSROW + c] = cc * a0 - sn * a1;
      Ub[i1 * LDSROW + c] = sn * a0 + cc * a1;
    }
    __syncthreads();

    if (q == 6) {                               // CZ layer after each depth
      for (int i = 16 * w; i < 16 * w + 16; ++i) {
        const int par = __popc(i & (i >> 1) & 0x3F) & 1;
        if (par) Ub[i * LDSROW + c] = -Ub[i * LDSROW + c];
      }
      __syncthreads();
    }
  }

  // write out row-major U[i][c]; coalesced across c
  for (int i = 16 * w; i < 16 * w + 16; ++i)
    Uout[i * DIMQ + c] = Ub[i * LDSROW + c];
}

// ---------------------------------------------------------------------------
// Kernel 2: C = emb @ U^T via V_WMMA_F32_16X16X4_F32, then probs -> Z readout.
// ---------------------------------------------------------------------------
__global__ void __launch_bounds__(256)
qnat_main(const float* __restrict__ A,      // BATCH x 128
          const float* __restrict__ U,      // 128 x 128 (row major: U[i][k])
          float* __restrict__ out) {        // BATCH x 7
  extern __shared__ float smem[];
  float* Us = smem;                          // 128 x LDSROW

  // Stage U into padded LDS (coalesced global reads).
  for (int idx = threadIdx.x; idx < DIMQ * DIMQ; idx += 256) {
    const int r = idx >> 7, cc = idx & 127;
    Us[r * LDSROW + cc] = U[idx];
  }
  __syncthreads();

  const int wave = threadIdx.x >> 5;
  const int lane = threadIdx.x & 31;
  const int l16  = lane & 15;
  const int h    = lane >> 4;                // which half of the wave
  float* Pw = smem + DIMQ * LDSROW + wave * (16 * LDSROW);  // per-wave P tile

  for (int t = 0; t < TPW; ++t) {
    const int rbase = blockIdx.x * ROWS_PER_BLOCK + (wave * TPW + t) * 16;
    const float* Arow = A + (size_t)(rbase + l16) * DIMQ + 2 * h;

    v8f acc[8] = {};
    float ssq = 0.0f;

    // K loop: 32 steps of K=4; 8 N-tiles -> 256 WMMAs per wave-tile.
#pragma unroll
    for (int kk = 0; kk < 32; ++kk) {
      const v2f af = *(const v2f*)(Arow + 4 * kk);          // A frag (16x4)
      ssq += af[0] * af[0] + af[1] * af[1];
#pragma unroll
      for (int j = 0; j < 8; ++j) {
        const v2f bf = *(const v2f*)(Us + (16 * j + l16) * LDSROW + 4 * kk + 2 * h);
        acc[j] = __builtin_amdgcn_wmma_f32_16x16x4_f32(
            false, af, false, bf, (short)0, acc[j], false, false);
      }
    }

    // Row ||s||^2: combine the two half-row partials, then broadcast the
    // inverse to the rows (M = r + 8h) this lane's C registers hold.
    ssq += __shfl_xor(ssq, 16, 32);
    const float invn = 1.0f / ssq;
    float invr[8];
#pragma unroll
    for (int r = 0; r < 8; ++r) invr[r] = __shfl(invn, r + 8 * h, 32);

    // P = C^2 / ||s||^2 -> per-wave LDS (32 lanes -> 32 distinct banks).
#pragma unroll
    for (int j = 0; j < 8; ++j) {
#pragma unroll
      for (int r = 0; r < 8; ++r) {
        const float cv = acc[j][r];
        Pw[(r + 8 * h) * LDSROW + 16 * j + l16] = cv * cv * invr[r];
      }
    }
    asm volatile("s_wait_dscnt 0x0" ::: "memory");   // order DS RAW in-wave

    // Readout: out[m] = S_total - 2 * S_{bit m set}. Lane handles row l16,
    // i in [64h, 64h+64): bits 0..5 compile-time, bit 6 == h.
    const float* Prow = Pw + l16 * LDSROW + 64 * h;
    float lt = 0.f, b0 = 0.f, b1 = 0.f, b2 = 0.f, b3 = 0.f, b4 = 0.f, b5 = 0.f;
#pragma unroll
    for (int i4 = 0; i4 < 16; ++i4) {
      const v4f p = *(const v4f*)(Prow + 4 * i4);
#pragma unroll
      for (int e = 0; e < 4; ++e) {
        const int il = 4 * i4 + e;               // compile-time
        const float pv = p[e];
        lt += pv;
        if (il & 1)  b0 += pv;
        if (il & 2)  b1 += pv;
        if (il & 4)  b2 += pv;
        if (il & 8)  b3 += pv;
        if (il & 16) b4 += pv;
        if (il & 32) b5 += pv;
      }
    }
    float b6 = h ? lt : 0.0f;
    lt += __shfl_xor(lt, 16, 32);
    b0 += __shfl_xor(b0, 16, 32);
    b1 += __shfl_xor(b1, 16, 32);
    b2 += __shfl_xor(b2, 16, 32);
    b3 += __shfl_xor(b3, 16, 32);
    b4 += __shfl_xor(b4, 16, 32);
    b5 += __shfl_xor(b5, 16, 32);
    b6 += __shfl_xor(b6, 16, 32);

    const float o0 = lt - 2.0f * b0, o1 = lt - 2.0f * b1, o2 = lt - 2.0f * b2;
    const float o3 = lt - 2.0f * b3, o4 = lt - 2.0f * b4, o5 = lt - 2.0f * b5;
    const float o6 = lt - 2.0f * b6;

    float* orow = out + (size_t)(rbase + l16) * 7;
    // lo lanes write m=0..3, hi lanes m=4..6 (register indices compile-time).
    orow[4 * h + 0] = h ? o4 : o0;
    orow[4 * h + 1] = h ? o5 : o1;
    orow[4 * h + 2] = h ? o6 : o2;
    if (!h) orow[3] = o3;
  }
}

extern "C" void kernel_launch(void* const* d_in, const int* in_sizes, int n_in,
                              void* d_out, int out_size, void* d_ws, size_t ws_size,
                              hipStream_t stream) {
  const float* emb   = (const float*)d_in[0];   // BATCH x 128
  const float* theta = (const float*)d_in[1];   // 14
  float* Umat = (float*)d_ws;                   // 128*128 f32 = 64 KB scratch
  float* outp = (float*)d_out;                  // BATCH x 7

  const int batch = in_sizes[0] / DIMQ;
  const int grid  = batch / ROWS_PER_BLOCK;     // 262144/256 = 1024

  const size_t ubytes = (size_t)(DIMQ * LDSROW) * sizeof(float);
  qnat_build_u<<<1, 1024, ubytes, stream>>>(theta, Umat);

  const size_t shbytes = (size_t)(DIMQ * LDSROW + WAVES * 16 * LDSROW) * sizeof(float);
  qnat_main<<<grid, 256, shbytes, stream>>>(emb, Umat, outp);
}